// SGConvolution_75900662054979
// MI455X (gfx1250) — compile-verified
//
#include <hip/hip_runtime.h>
#include <hip/hip_bf16.h>
#include <stdint.h>

// ---------------------------------------------------------------------------
// SGConv (order=2):  out = A @ (A @ x),  A sparse COO (E edges), x [N,32] f32.
// Strategy: build dst-CSR once (deterministic 2-level scan), then two
// atomic-free gather SpMM passes, one wave32 per row, lane == feature.
// Edge (src,w) pairs are staged into LDS via gfx1250 async-to-LDS loads.
// ---------------------------------------------------------------------------

#define TPB 256

__global__ void k_zero_i32(int* __restrict__ p, int n) {
    int i = blockIdx.x * TPB + threadIdx.x;
    if (i < n) p[i] = 0;
}

__global__ void k_zero_f32(float* __restrict__ p, int n) {
    int i = blockIdx.x * TPB + threadIdx.x;
    if (i < n) p[i] = 0.0f;
}

// counts[dst]++ for every edge
__global__ void k_count(const int* __restrict__ edst, int* __restrict__ cnt, int E) {
    int e = blockIdx.x * TPB + threadIdx.x;
    if (e < E) atomicAdd(&cnt[edst[e]], 1);
}

// per-block exclusive scan of cnt -> local (in row_start), block totals -> bsum
__global__ void k_scanA(const int* __restrict__ cnt, int* __restrict__ local,
                        int* __restrict__ bsum, int n) {
    __shared__ int sh[TPB];
    int i = blockIdx.x * TPB + threadIdx.x;
    int v = (i < n) ? cnt[i] : 0;
    sh[threadIdx.x] = v;
    __syncthreads();
    for (int off = 1; off < TPB; off <<= 1) {
        int y = (threadIdx.x >= off) ? sh[threadIdx.x - off] : 0;
        __syncthreads();
        sh[threadIdx.x] += y;
        __syncthreads();
    }
    int incl = sh[threadIdx.x];
    if (i < n) local[i] = incl - v;                 // exclusive within block
    if (threadIdx.x == TPB - 1) bsum[blockIdx.x] = incl;
}

// single-block exclusive scan (in place) over nb block sums, with carry loop
__global__ void k_scanB(int* __restrict__ bsum, int nb) {
    __shared__ int sh[TPB];
    int carry = 0;
    for (int base = 0; base < nb; base += TPB) {
        int i = base + threadIdx.x;
        int v = (i < nb) ? bsum[i] : 0;
        sh[threadIdx.x] = v;
        __syncthreads();
        for (int off = 1; off < TPB; off <<= 1) {
            int y = (threadIdx.x >= off) ? sh[threadIdx.x - off] : 0;
            __syncthreads();
            sh[threadIdx.x] += y;
            __syncthreads();
        }
        int incl = sh[threadIdx.x];
        if (i < nb) bsum[i] = (incl - v) + carry;   // exclusive + carry
        __syncthreads();
        carry += sh[TPB - 1];
        __syncthreads();
    }
}

// row_start[i] = local[i] + bsum[block(i)]; cursor[i] = row_start[i]
__global__ void k_scanC(int* __restrict__ row_start, int* __restrict__ cursor,
                        const int* __restrict__ bsum, int n) {
    int i = blockIdx.x * TPB + threadIdx.x;
    if (i < n) {
        int rs = row_start[i] + bsum[i >> 8];       // 256 == TPB
        row_start[i] = rs;
        cursor[i]    = rs;
    }
}

// scatter edges into CSR pool as packed (src, w) 8-byte pairs
__global__ void k_fill(const int* __restrict__ esrc, const int* __restrict__ edst,
                       const float* __restrict__ ew,
                       int* __restrict__ cursor,
                       unsigned long long* __restrict__ pairs, int E) {
    int e = blockIdx.x * TPB + threadIdx.x;
    if (e < E) {
        int dst = edst[e];
        int pos = atomicAdd(&cursor[dst], 1);
        unsigned int wb = __float_as_uint(ew[e]);
        pairs[pos] = ((unsigned long long)wb << 32) | (unsigned int)esrc[e];
    }
}

// one wave32 per row, lane == feature; edge pairs staged via async-to-LDS
__global__ void k_spmm(const unsigned long long* __restrict__ pairs,
                       const int* __restrict__ row_start,
                       const int* __restrict__ row_cnt,
                       const float* __restrict__ xin,
                       float* __restrict__ xout, int nrows) {
    __shared__ unsigned long long stage[TPB / 32][32];   // 8 waves * 32 pairs * 8B
    int t    = blockIdx.x * TPB + threadIdx.x;
    int row  = t >> 5;
    int lane = threadIdx.x & 31;
    int wv   = threadIdx.x >> 5;
    if (row >= nrows) return;

    int start = row_start[row];
    int cnt   = row_cnt[row];
    float acc = 0.0f;

    // LDS byte offset of this wave's staging slot (low 32 bits of flat ptr)
    unsigned int lds_base = (unsigned int)(uintptr_t)(&stage[wv][0]);

    for (int c = 0; c < cnt; c += 32) {
        int m = cnt - c;
        if (m > 32) m = 32;
        // make sure prior chunk's ds reads are done before overwriting the slot
        asm volatile("s_wait_dscnt 0x0" ::: "memory");
        if (lane < m) {
            unsigned int       ldsoff = lds_base + (unsigned int)lane * 8u;
            unsigned long long gaddr  =
                (unsigned long long)(uintptr_t)(pairs + start + c + lane);
            asm volatile("global_load_async_to_lds_b64 %0, %1, off"
                         :: "v"(ldsoff), "v"(gaddr) : "memory");
        }
        asm volatile("s_wait_asynccnt 0x0" ::: "memory");
        #pragma unroll 4
        for (int j = 0; j < m; ++j) {
            unsigned long long p = stage[wv][j];          // uniform ds_load_b64
            int          src = (int)(unsigned int)(p & 0xffffffffull);
            float        w   = __uint_as_float((unsigned int)(p >> 32));
            acc = fmaf(w, xin[src * 32 + lane], acc);     // coalesced 128B gather
        }
    }
    xout[row * 32 + lane] = acc;
}

// ---- fallback path (workspace too small for CSR): atomic scatter-add ------
__global__ void k_scatter(const int* __restrict__ esrc, const int* __restrict__ edst,
                          const float* __restrict__ ew,
                          const float* __restrict__ xin,
                          float* __restrict__ xout, int E) {
    int t    = blockIdx.x * TPB + threadIdx.x;
    int e    = t >> 5;
    int lane = threadIdx.x & 31;
    if (e < E) {
        int   src = esrc[e];
        int   dst = edst[e];
        float w   = ew[e];
        atomicAdd(&xout[dst * 32 + lane], w * xin[src * 32 + lane]);
    }
}

static inline size_t align_up(size_t v, size_t a) { return (v + a - 1) & ~(a - 1); }

extern "C" void kernel_launch(void* const* d_in, const int* in_sizes, int n_in,
                              void* d_out, int out_size, void* d_ws, size_t ws_size,
                              hipStream_t stream) {
    const float* x    = (const float*)d_in[0];
    const int*   esrc = (const int*)  d_in[1];
    const int*   edst = (const int*)  d_in[2];
    const float* ew   = (const float*)d_in[3];
    float*       out  = (float*)d_out;

    const int N = in_sizes[0] / 32;
    const int E = in_sizes[1];
    const int nb = (N + TPB - 1) / TPB;

    char* ws = (char*)d_ws;

    // workspace layout (CSR path)
    size_t o_bsum  = 0;
    size_t o_cnt   = align_up(o_bsum + (size_t)nb * 4, 64);
    size_t o_start = align_up(o_cnt  + (size_t)N  * 4, 64);
    size_t o_cur   = align_up(o_start+ (size_t)N  * 4, 64);
    size_t o_pairs = align_up(o_cur  + (size_t)N  * 4, 64);
    size_t o_y     = align_up(o_pairs+ (size_t)E  * 8, 64);
    size_t need    = o_y + (size_t)N * 32 * 4;

    const int gN   = (N + TPB - 1) / TPB;            // N threads
    const int gE   = (E + TPB - 1) / TPB;            // E threads
    const int gRow = (N * 32 + TPB - 1) / TPB;       // wave per row
    const int gEd  = ((size_t)E * 32 + TPB - 1) / TPB;

    if (ws_size >= need) {
        int* bsum      = (int*)(ws + o_bsum);
        int* row_cnt   = (int*)(ws + o_cnt);
        int* row_start = (int*)(ws + o_start);
        int* cursor    = (int*)(ws + o_cur);
        unsigned long long* pairs = (unsigned long long*)(ws + o_pairs);
        float* y       = (float*)(ws + o_y);

        k_zero_i32<<<gN, TPB, 0, stream>>>(row_cnt, N);
        k_count  <<<gE, TPB, 0, stream>>>(edst, row_cnt, E);
        k_scanA  <<<nb, TPB, 0, stream>>>(row_cnt, row_start, bsum, N);
        k_scanB  <<<1,  TPB, 0, stream>>>(bsum, nb);
        k_scanC  <<<gN, TPB, 0, stream>>>(row_start, cursor, bsum, N);
        k_fill   <<<gE, TPB, 0, stream>>>(esrc, edst, ew, cursor, pairs, E);

        k_spmm<<<gRow, TPB, 0, stream>>>(pairs, row_start, row_cnt, x,   y, N);
        k_spmm<<<gRow, TPB, 0, stream>>>(pairs, row_start, row_cnt, y, out, N);
    } else {
        // fallback: two atomic scatter passes; needs only y = N*128 bytes
        float* y = (float*)ws;
        k_zero_f32<<<gRow, TPB, 0, stream>>>(y, N * 32);
        k_scatter <<<gEd, TPB, 0, stream>>>(esrc, edst, ew, x, y, E);
        k_zero_f32<<<gRow, TPB, 0, stream>>>(out, N * 32);
        k_scatter <<<gEd, TPB, 0, stream>>>(esrc, edst, ew, y, out, E);
    }
}